// NodeTransformer_6502580486379
// MI455X (gfx1250) — compile-verified
//
#include <hip/hip_runtime.h>

// ---------------------------------------------------------------------------
// 2-layer PyG TransformerConv (heads=1) + skip MLP for MI455X (gfx1250).
//  - fp32 WMMA (v_wmma_f32_16x16x4_f32) for all dense GEMMs (GEMMs are a small
//    fraction of runtime -> keep full fp32 fidelity, still use the matrix pipe)
//  - edge phase: L2-resident gather + native f32 atomic scatter (node arrays
//    are 12.8MB each << 192MB L2)
//  - streamed one-touch edge data (edge_attr / edge_index / score) uses
//    non-temporal loads/stores so it does not evict the L2-resident node set
// Workspace: ~93 MB carved from d_ws.
// ---------------------------------------------------------------------------

#define NN 50000            // nodes
#define EE 800000           // edges
#define HH 64               // hidden channels
#define NM (NN * HH)        // node-matrix element count

typedef __attribute__((ext_vector_type(2))) float v2f;
typedef __attribute__((ext_vector_type(8))) float v8f;

// ---- WMMA helpers ----------------------------------------------------------
// One wave computes one 16x16 f32 tile of C += A(16xK) * B(KxN)[:, col0:col0+16]
// A frag (16x4 f32): lanes 0-15 hold K={k0,k0+1}, lanes 16-31 hold K={k0+2,k0+3}
// B frag mirrored; C/D: VGPR r -> row r (lanes 0-15) / row r+8 (lanes 16-31).
__device__ inline v8f wmma_f32_tile(const float* __restrict__ A, int lda,
                                    const float* __restrict__ B, int ldb,
                                    int col0, int K, int lane, v8f c) {
  const int m  = lane & 15;
  const int ks = (lane >> 4) << 1;      // 0 for lanes 0-15, 2 for lanes 16-31
  const int n  = lane & 15;
  for (int k0 = 0; k0 < K; k0 += 4) {
    v2f a, b;
    const float* ap = A + m * lda + k0 + ks;
    a.x = ap[0];
    a.y = ap[1];
    b.x = B[(k0 + ks) * ldb + col0 + n];
    b.y = B[(k0 + ks + 1) * ldb + col0 + n];
    c = __builtin_amdgcn_wmma_f32_16x16x4_f32(false, a, false, b, (short)0, c,
                                              false, false);
  }
  return c;
}

__device__ inline void store_tile(float* __restrict__ C, int ldc, int row0,
                                  int col0, v8f c, int lane,
                                  const float* __restrict__ bias, bool relu) {
  const int col   = col0 + (lane & 15);
  const int rbase = row0 + ((lane >> 4) << 3);   // lanes 16-31 -> rows +8
  const float bv  = bias ? bias[col] : 0.f;
#pragma unroll
  for (int r = 0; r < 8; ++r) {
    float val = c[r] + bv;
    if (relu && val < 0.f) val = 0.f;
    C[(rbase + r) * ldc + col] = val;
  }
}

// ---- order-preserving float<->uint map for atomicMax ----------------------
__device__ inline unsigned fenc(float f) {
  unsigned u = __float_as_uint(f);
  return (u & 0x80000000u) ? ~u : (u | 0x80000000u);
}
__device__ inline float fdec(unsigned u) {
  return (u & 0x80000000u) ? __uint_as_float(u & 0x7fffffffu)
                           : __uint_as_float(~u);
}

// ---------------------------------------------------------------------------
// Kernel A: q = xWq+bq, k = xWk+bk, v = xWv+bv, skip = xWs+bs  (all [N,64])
// grid = (N/16, 4 mats), block = 128 (4 waves = 4 column tiles of 16).
// ---------------------------------------------------------------------------
__global__ __launch_bounds__(128) void node_linear4_kernel(
    const float* __restrict__ x,
    const float* __restrict__ wq, const float* __restrict__ bq, float* __restrict__ q,
    const float* __restrict__ wk, const float* __restrict__ bk, float* __restrict__ k,
    const float* __restrict__ wv, const float* __restrict__ bv, float* __restrict__ v,
    const float* __restrict__ ws, const float* __restrict__ bs, float* __restrict__ s) {
  const int lane = threadIdx.x & 31;
  const int wid  = threadIdx.x >> 5;
  const int row0 = blockIdx.x * 16;
  const float* W;
  const float* B;
  float* O;
  switch (blockIdx.y) {
    case 0:  W = wq; B = bq; O = q; break;
    case 1:  W = wk; B = bk; O = k; break;
    case 2:  W = wv; B = bv; O = v; break;
    default: W = ws; B = bs; O = s; break;
  }
  v8f c = {};
  c = wmma_f32_tile(x + row0 * HH, HH, W, HH, wid * 16, HH, lane, c);
  store_tile(O, HH, row0, wid * 16, c, lane, B, false);
}

// ---------------------------------------------------------------------------
// Edge pass 1: e = edge_attr@We (on the fly), score = <q[dst], k[src]+e>/8,
// store score (NT), atomicMax per-dst running max. One wave per edge,
// lane l owns channels {2l, 2l+1}. We (32x64 = 8KB) staged in LDS.
// Streamed edge data (ei / ea / score) is non-temporal: keeps q/k resident
// in L2 for the random gathers.
// ---------------------------------------------------------------------------
__global__ __launch_bounds__(256) void edge_score_kernel(
    const long long* __restrict__ ei, const float* __restrict__ ea,
    const float* __restrict__ we, const float* __restrict__ q,
    const float* __restrict__ k, float* __restrict__ score,
    unsigned* __restrict__ smax) {
  __shared__ float wel[32 * HH];
  for (int i = threadIdx.x; i < 32 * HH; i += blockDim.x) wel[i] = we[i];
  __syncthreads();

  const int lane = threadIdx.x & 31;
  const int eid  = blockIdx.x * (blockDim.x >> 5) + (threadIdx.x >> 5);
  const int src  = (int)__builtin_nontemporal_load(ei + eid);
  const int dst  = (int)__builtin_nontemporal_load(ei + EE + eid);

  const float ear = __builtin_nontemporal_load(ea + eid * 32 + lane);
  const int   h0  = lane * 2;
  float e0 = 0.f, e1 = 0.f;
#pragma unroll
  for (int d = 0; d < 32; ++d) {
    const float  sdv = __shfl(ear, d, 32);
    const float2 w2  = *(const float2*)&wel[d * HH + h0];   // one ds_load_b64
    e0 = fmaf(sdv, w2.x, e0);
    e1 = fmaf(sdv, w2.y, e1);
  }
  const float2 kk = *(const float2*)(k + src * HH + h0);
  const float2 qq = *(const float2*)(q + dst * HH + h0);
  float p = qq.x * (kk.x + e0) + qq.y * (kk.y + e1);
#pragma unroll
  for (int off = 16; off; off >>= 1) p += __shfl_xor(p, off, 32);
  if (lane == 0) {
    const float sc = p * 0.125f;           // 1/sqrt(64)
    __builtin_nontemporal_store(sc, score + eid);
    atomicMax(&smax[dst], fenc(sc));
  }
}

// ---------------------------------------------------------------------------
// Edge pass 2: a = exp(score - smax[dst]); denom[dst]+=a;
// acc[dst] += a*(v[src]+e). Native f32 atomics into L2-resident acc.
// ---------------------------------------------------------------------------
__global__ __launch_bounds__(256) void edge_scatter_kernel(
    const long long* __restrict__ ei, const float* __restrict__ ea,
    const float* __restrict__ we, const float* __restrict__ v,
    const float* __restrict__ score, const unsigned* __restrict__ smax,
    float* __restrict__ acc, float* __restrict__ denom) {
  __shared__ float wel[32 * HH];
  for (int i = threadIdx.x; i < 32 * HH; i += blockDim.x) wel[i] = we[i];
  __syncthreads();

  const int lane = threadIdx.x & 31;
  const int eid  = blockIdx.x * (blockDim.x >> 5) + (threadIdx.x >> 5);
  const int src  = (int)__builtin_nontemporal_load(ei + eid);
  const int dst  = (int)__builtin_nontemporal_load(ei + EE + eid);

  const float ear = __builtin_nontemporal_load(ea + eid * 32 + lane);
  const int   h0  = lane * 2;
  float e0 = 0.f, e1 = 0.f;
#pragma unroll
  for (int d = 0; d < 32; ++d) {
    const float  sdv = __shfl(ear, d, 32);
    const float2 w2  = *(const float2*)&wel[d * HH + h0];
    e0 = fmaf(sdv, w2.x, e0);
    e1 = fmaf(sdv, w2.y, e1);
  }
  const float sc = __builtin_nontemporal_load(score + eid);
  const float a  = __expf(sc - fdec(smax[dst]));
  if (lane == 0) unsafeAtomicAdd(&denom[dst], a);
  const float2 vv = *(const float2*)(v + src * HH + h0);
  unsafeAtomicAdd(&acc[dst * HH + h0],     a * (vv.x + e0));
  unsafeAtomicAdd(&acc[dst * HH + h0 + 1], a * (vv.y + e1));
}

// ---------------------------------------------------------------------------
// Node finalize: h = relu(acc/denom + (xWs+bs)). denom==0 -> no messages.
// acc / skip are read-once streams -> non-temporal loads.
// ---------------------------------------------------------------------------
__global__ __launch_bounds__(256) void node_finalize_kernel(
    const float* __restrict__ acc, const float* __restrict__ denom,
    const float* __restrict__ skipb, float* __restrict__ h) {
  const int i = blockIdx.x * blockDim.x + threadIdx.x;
  if (i >= NM) return;
  const float d = denom[i >> 6];
  const float av = __builtin_nontemporal_load(acc + i);
  const float sv = __builtin_nontemporal_load(skipb + i);
  const float m = (d > 0.f) ? (av / d) : 0.f;
  const float r = m + sv;
  h[i] = (r > 0.f) ? r : 0.f;
}

// ---------------------------------------------------------------------------
// Final: out = relu([h1 h2] @ skip_w + skip_b), skip_w is [128,64] row-major.
// ---------------------------------------------------------------------------
__global__ __launch_bounds__(128) void skip_linear_kernel(
    const float* __restrict__ h1, const float* __restrict__ h2,
    const float* __restrict__ w, const float* __restrict__ b,
    float* __restrict__ out) {
  const int lane = threadIdx.x & 31;
  const int wid  = threadIdx.x >> 5;
  const int row0 = blockIdx.x * 16;
  const int col0 = wid * 16;
  v8f c = {};
  c = wmma_f32_tile(h1 + row0 * HH, HH, w, HH, col0, HH, lane, c);            // K rows 0..63
  c = wmma_f32_tile(h2 + row0 * HH, HH, w + HH * HH, HH, col0, HH, lane, c);  // K rows 64..127
  store_tile(out, HH, row0, col0, c, lane, b, true);
}

// ---------------------------------------------------------------------------
extern "C" void kernel_launch(void* const* d_in, const int* in_sizes, int n_in,
                              void* d_out, int out_size, void* d_ws,
                              size_t ws_size, hipStream_t stream) {
  const float*     x   = (const float*)d_in[0];
  const long long* ei  = (const long long*)d_in[1];   // int64 edge_index [2,E]
  const float*     ea  = (const float*)d_in[2];
  const float* q1w = (const float*)d_in[3];  const float* q1b = (const float*)d_in[4];
  const float* k1w = (const float*)d_in[5];  const float* k1b = (const float*)d_in[6];
  const float* v1w = (const float*)d_in[7];  const float* v1b = (const float*)d_in[8];
  const float* e1w = (const float*)d_in[9];
  const float* s1w = (const float*)d_in[10]; const float* s1b = (const float*)d_in[11];
  const float* q2w = (const float*)d_in[12]; const float* q2b = (const float*)d_in[13];
  const float* k2w = (const float*)d_in[14]; const float* k2b = (const float*)d_in[15];
  const float* v2w = (const float*)d_in[16]; const float* v2b = (const float*)d_in[17];
  const float* e2w = (const float*)d_in[18];
  const float* s2w = (const float*)d_in[19]; const float* s2b = (const float*)d_in[20];
  const float* skw = (const float*)d_in[21]; const float* skb = (const float*)d_in[22];

  // Workspace carve (floats): 7*NM + EE + 2*NN  ~= 93 MB
  float* p = (float*)d_ws;
  float* qb    = p; p += NM;
  float* kb    = p; p += NM;
  float* vb    = p; p += NM;
  float* skp   = p; p += NM;
  float* acc   = p; p += NM;
  float* h1    = p; p += NM;
  float* h2    = p; p += NM;
  float* score = p; p += EE;
  float* denom = p; p += NN;
  unsigned* smax = (unsigned*)p;

  const dim3 gN(NN / 16, 4);
  const dim3 bN(128);
  const int  gE = EE / 8;   // 8 waves (edges) per 256-thread block
  const int  bE = 256;

  for (int layer = 0; layer < 2; ++layer) {
    hipMemsetAsync(acc,   0, (size_t)NM * sizeof(float), stream);
    hipMemsetAsync(denom, 0, (size_t)NN * sizeof(float), stream);
    hipMemsetAsync(smax,  0, (size_t)NN * sizeof(unsigned), stream);  // 0 < fenc(any)

    const float* xin = layer ? h1 : x;
    const float* WQ = layer ? q2w : q1w; const float* BQ = layer ? q2b : q1b;
    const float* WK = layer ? k2w : k1w; const float* BK = layer ? k2b : k1b;
    const float* WV = layer ? v2w : v1w; const float* BV = layer ? v2b : v1b;
    const float* WS = layer ? s2w : s1w; const float* BS = layer ? s2b : s1b;
    const float* WE = layer ? e2w : e1w;

    node_linear4_kernel<<<gN, bN, 0, stream>>>(xin, WQ, BQ, qb, WK, BK, kb,
                                               WV, BV, vb, WS, BS, skp);
    edge_score_kernel<<<gE, bE, 0, stream>>>(ei, ea, WE, qb, kb, score, smax);
    edge_scatter_kernel<<<gE, bE, 0, stream>>>(ei, ea, WE, vb, score, smax,
                                               acc, denom);
    float* hout = layer ? h2 : h1;
    node_finalize_kernel<<<(NM + 255) / 256, 256, 0, stream>>>(acc, denom, skp,
                                                               hout);
  }
  skip_linear_kernel<<<dim3(NN / 16), bN, 0, stream>>>(h1, h2, skw, skb,
                                                       (float*)d_out);
}